// Graph_64055142252969
// MI455X (gfx1250) — compile-verified
//
#include <hip/hip_runtime.h>
#include <hip/hip_bf16.h>

typedef __attribute__((ext_vector_type(2))) float v2f;
typedef __attribute__((ext_vector_type(8))) float v8f;

// ---------------------------------------------------------------------------
// Zero-fill
// ---------------------------------------------------------------------------
__global__ void fill_zero_kernel(float* __restrict__ p, long long n) {
    long long i = (long long)blockIdx.x * blockDim.x + threadIdx.x;
    if (i < n) p[i] = 0.0f;
}

// ---------------------------------------------------------------------------
// Degree count: cnt[dst] += 1 for each edge
// ---------------------------------------------------------------------------
__global__ void degree_kernel(const int* __restrict__ dst, float* __restrict__ cnt,
                              long long nEdge) {
    long long e = (long long)blockIdx.x * blockDim.x + threadIdx.x;
    if (e < nEdge) atomicAdd(&cnt[dst[e]], 1.0f);
}

// inv[i] = 1 / max(cnt[i], 1)   (in place over the count buffer)
__global__ void recip_kernel(float* __restrict__ cnt, int n) {
    int i = blockIdx.x * blockDim.x + threadIdx.x;
    if (i < n) cnt[i] = 1.0f / fmaxf(cnt[i], 1.0f);
}

// ---------------------------------------------------------------------------
// Pack W[K=128][OUT] into WMMA B-fragment order:
//   Wp[((k0q*2 + half)*OUT + col)*2 + rem] = W[(4*k0q + 2*half + rem)][col]
// so a lane's B fragment (W[k][col], W[k+1][col]) is one aligned b64 load.
// ---------------------------------------------------------------------------
__global__ void pack_w_kernel(const float* __restrict__ W, float* __restrict__ Wp,
                              int OUT) {
    int i = blockIdx.x * blockDim.x + threadIdx.x;
    if (i >= 128 * OUT) return;
    int k = i / OUT, c = i - k * OUT;
    int k0q  = k >> 2;
    int half = (k >> 1) & 1;
    int rem  = k & 1;
    Wp[(((k0q * 2 + half) * (long long)OUT) + c) * 2 + rem] = W[i];
}

// ---------------------------------------------------------------------------
// Edge scatter: summed[dst] += w_e * feat[src]
// 32 lanes per edge, 4 consecutive floats per lane (D = 128)
// ---------------------------------------------------------------------------
__global__ void __launch_bounds__(256)
scatter_kernel(const float* __restrict__ feat,
               const int* __restrict__ src, const int* __restrict__ dst,
               const float* __restrict__ w, float* __restrict__ summed,
               long long nEdge) {
    long long tid = (long long)blockIdx.x * 256 + threadIdx.x;
    long long e = tid >> 5;
    if (e >= nEdge) return;
    const int base = (threadIdx.x & 31) * 4;
    const int s = src[e];
    const int d = dst[e];
    const float ww = w[e];
    const float4 v = *(const float4*)(feat + (long long)s * 128 + base);
    float* p = summed + (long long)d * 128 + base;
    atomicAdd(p + 0, v.x * ww);
    atomicAdd(p + 1, v.y * ww);
    atomicAdd(p + 2, v.z * ww);
    atomicAdd(p + 3, v.w * ww);
}

// ---------------------------------------------------------------------------
// Fused GraphConv GEMM (fp32 WMMA 16x16x4):
//   Out[r, c] = sum_k S[r,k]*inv[r]*WpRel[k,c] + sum_k X[r,k]*WpRoot[k,c] + bias[c]
// WpRel/WpRoot are in packed B-fragment layout (see pack_w_kernel).
// One wave computes a 16-row x 128-col slice (8 accumulator tiles).
// Block = 256 threads = 8 waves = 128 rows; grid.y = 128-wide column group.
// ---------------------------------------------------------------------------
__global__ void __launch_bounds__(256)
gemm_graphconv_kernel(const float* __restrict__ S, const float* __restrict__ inv,
                      const float* __restrict__ X,
                      const float* __restrict__ WpRel,
                      const float* __restrict__ WpRoot,
                      const float* __restrict__ bias,
                      float* __restrict__ Out, int nRows, int ldOut) {
    const int wave    = threadIdx.x >> 5;
    const int lane    = threadIdx.x & 31;
    const int rowBase = blockIdx.x * 128 + wave * 16;
    if (rowBase >= nRows) return;   // whole-wave exit: EXEC stays all-ones for WMMA
    const int colBase = blockIdx.y * 128;
    const int l15     = lane & 15;
    const int half    = lane >> 4;           // 0: K pair {0,1}; 1: K pair {2,3}

    // A fragment: row = rowBase + l15, contiguous K pair at k0 + 2*half
    const int   aRow  = rowBase + l15;
    const float scale = inv[aRow];
    const float* aS   = S + (long long)aRow * 128 + 2 * half;
    const float* aX   = X + (long long)aRow * 128 + 2 * half;

    // Packed-W base (v2f units): index = (k0q*2 + half)*ldOut + col
    const v2f* wRel  = (const v2f*)WpRel  + (long long)half * ldOut + colBase + l15;
    const v2f* wRoot = (const v2f*)WpRoot + (long long)half * ldOut + colBase + l15;

    v8f acc[8];
#pragma unroll
    for (int t = 0; t < 8; ++t) {
        const float bv = bias[colBase + t * 16 + l15];
#pragma unroll
        for (int v = 0; v < 8; ++v) acc[t][v] = bv;
    }

    // ---- mean @ Wrel : A = S[row, k] * inv[row] ----
    for (int k0 = 0; k0 < 128; k0 += 4) {
        v2f a = *(const v2f*)(aS + k0);
        a *= scale;
        const v2f* wp = wRel + (long long)(k0 >> 2) * 2 * ldOut;
        v2f b[8];
#pragma unroll
        for (int t = 0; t < 8; ++t) b[t] = wp[t * 16];   // 8 coalesced b64 loads
#pragma unroll
        for (int t = 0; t < 8; ++t)
            acc[t] = __builtin_amdgcn_wmma_f32_16x16x4_f32(
                false, a, false, b[t], (short)0, acc[t], false, false);
    }

    // ---- x @ Wroot ----
    for (int k0 = 0; k0 < 128; k0 += 4) {
        v2f a = *(const v2f*)(aX + k0);
        const v2f* wp = wRoot + (long long)(k0 >> 2) * 2 * ldOut;
        v2f b[8];
#pragma unroll
        for (int t = 0; t < 8; ++t) b[t] = wp[t * 16];
#pragma unroll
        for (int t = 0; t < 8; ++t)
            acc[t] = __builtin_amdgcn_wmma_f32_16x16x4_f32(
                false, a, false, b[t], (short)0, acc[t], false, false);
    }

    // ---- store C/D: VGPR v -> row rowBase + v + 8*half, col = colBase+t*16+l15
#pragma unroll
    for (int t = 0; t < 8; ++t) {
        const int col = colBase + t * 16 + l15;
#pragma unroll
        for (int v = 0; v < 8; ++v) {
            const int row = rowBase + v + 8 * half;
            Out[(long long)row * ldOut + col] = acc[t][v];
        }
    }
}

// ---------------------------------------------------------------------------
// Launch
// ---------------------------------------------------------------------------
extern "C" void kernel_launch(void* const* d_in, const int* in_sizes, int n_in,
                              void* d_out, int out_size, void* d_ws, size_t ws_size,
                              hipStream_t stream) {
    const float* x      = (const float*)d_in[0];
    const int*   eidx   = (const int*)d_in[1];   // edge_index flattened [2, E]
    const float* ew     = (const float*)d_in[2];
    const float* Wrel1  = (const float*)d_in[3];
    const float* b1     = (const float*)d_in[4];
    const float* Wroot1 = (const float*)d_in[5];
    const float* Wrel2  = (const float*)d_in[6];
    const float* b2     = (const float*)d_in[7];
    const float* Wroot2 = (const float*)d_in[8];

    const int N = in_sizes[0] / 128;
    const long long E = in_sizes[1] / 2;
    const int* src = eidx;
    const int* dst = eidx + E;

    // workspace layout:
    //   inv[N] | summed[N*128] | h[N*128] | Wp1rel[16K] | Wp1root[16K]
    //   | Wp2rel[32K] | Wp2root[32K]
    float* inv     = (float*)d_ws;
    float* summed  = inv + N;
    float* h       = summed + (size_t)N * 128;
    float* Wp1rel  = h + (size_t)N * 128;
    float* Wp1root = Wp1rel + 128 * 128;
    float* Wp2rel  = Wp1root + 128 * 128;
    float* Wp2root = Wp2rel + 128 * 256;
    float* out     = (float*)d_out;

    const int T = 256;
    const long long nSum = (long long)N * 128;

    // --- pack weights into B-fragment layout ---
    pack_w_kernel<<<dim3((128 * 128 + T - 1) / T), T, 0, stream>>>(Wrel1, Wp1rel, 128);
    pack_w_kernel<<<dim3((128 * 128 + T - 1) / T), T, 0, stream>>>(Wroot1, Wp1root, 128);
    pack_w_kernel<<<dim3((128 * 256 + T - 1) / T), T, 0, stream>>>(Wrel2, Wp2rel, 256);
    pack_w_kernel<<<dim3((128 * 256 + T - 1) / T), T, 0, stream>>>(Wroot2, Wp2root, 256);

    // --- degree + reciprocal (shared by both layers) ---
    {
        long long nz = (long long)N + nSum;  // inv + summed are adjacent
        fill_zero_kernel<<<dim3((unsigned)((nz + T - 1) / T)), T, 0, stream>>>(inv, nz);
        degree_kernel<<<dim3((unsigned)((E + T - 1) / T)), T, 0, stream>>>(dst, inv, E);
        recip_kernel<<<dim3((N + T - 1) / T), T, 0, stream>>>(inv, N);
    }

    // --- layer 1: scatter(x) -> summed; h = mean@Wrel1 + x@Wroot1 + b1 ---
    {
        long long nThreads = E * 32;
        scatter_kernel<<<dim3((unsigned)((nThreads + T - 1) / T)), T, 0, stream>>>(
            x, src, dst, ew, summed, E);
        gemm_graphconv_kernel<<<dim3((N + 127) / 128, 1), T, 0, stream>>>(
            summed, inv, x, Wp1rel, Wp1root, b1, h, N, 128);
    }

    // --- layer 2: scatter(h) -> summed; out = mean@Wrel2 + h@Wroot2 + b2 ---
    {
        fill_zero_kernel<<<dim3((unsigned)((nSum + T - 1) / T)), T, 0, stream>>>(summed, nSum);
        long long nThreads = E * 32;
        scatter_kernel<<<dim3((unsigned)((nThreads + T - 1) / T)), T, 0, stream>>>(
            h, src, dst, ew, summed, E);
        gemm_graphconv_kernel<<<dim3((N + 127) / 128, 2), T, 0, stream>>>(
            summed, inv, h, Wp2rel, Wp2root, b2, out, N, 256);
    }
}